// HybridModel_65481071395610
// MI455X (gfx1250) — compile-verified
//
#include <hip/hip_runtime.h>

#define HW 128
#define OW 127
#define IMG_ELEMS (HW * HW)     // 16384 floats = 64 KB
#define OUT_ELEMS (OW * OW)     // 16129
#define INV_OUT_ELEMS (1.0f / 16129.0f)
#define GAMMA_F 1e-4f
#define NLAYER 4

typedef __attribute__((ext_vector_type(2))) float v2f;
typedef __attribute__((ext_vector_type(8))) float v8f;
typedef __attribute__((ext_vector_type(4))) int   v4i;

#define AS_GLOBAL __attribute__((address_space(1)))
#define AS_LDS    __attribute__((address_space(3)))

// One block per image. 256 threads = 8 wave32s. Stage image in LDS (read HBM
// exactly once, via async global->LDS DMA when available), conv2x2 + sigmoid
// + partial sums, WMMA cross-lane reduce, write mean.
__global__ __launch_bounds__(256) void conv_sig_mean_kernel(
    const float* __restrict__ data, const float* __restrict__ conv_w,
    const float* __restrict__ conv_b, float* __restrict__ means)
{
    __shared__ float s_img[IMG_ELEMS];   // exactly 64 KB

    const int b   = blockIdx.x;
    const int tid = threadIdx.x;

    const float* src_f = data + (size_t)b * IMG_ELEMS;

#if __has_builtin(__builtin_amdgcn_global_load_async_to_lds_b128) && \
    __has_builtin(__builtin_amdgcn_s_wait_asynccnt)
    // ---- CDNA5 async DMA: global -> LDS, no VGPR staging (ASYNCcnt) ----
    {
        AS_GLOBAL v4i* g = (AS_GLOBAL v4i*)src_f;
        AS_LDS    v4i* l = (AS_LDS    v4i*)s_img;
#pragma unroll
        for (int i = 0; i < IMG_ELEMS / 4 / 256; ++i) {
            int idx = i * 256 + tid;            // 16 B per lane per issue
            __builtin_amdgcn_global_load_async_to_lds_b128(
                g + idx, l + idx, /*offset=*/0, /*cpol=*/0);
        }
        __builtin_amdgcn_s_wait_asynccnt(0);    // this wave's DMAs landed
    }
#else
    // ---- fallback: b128 loads through VGPRs ----
    {
        const float4* __restrict__ src4 = (const float4*)src_f;
        float4* dst4 = (float4*)s_img;
#pragma unroll
        for (int i = 0; i < IMG_ELEMS / 4 / 256; ++i)
            dst4[i * 256 + tid] = src4[i * 256 + tid];
    }
#endif

    const float w00 = conv_w[0], w01 = conv_w[1];
    const float w10 = conv_w[2], w11 = conv_w[3];
    const float bias = conv_b[0];

    __syncthreads();   // every wave waited on its own asyncs before arriving

    const int wave = tid >> 5;
    const int lane = tid & 31;

    // ---- conv 2x2 + sigmoid, per-thread partial sum ----
    float sum = 0.0f;
    for (int r = wave; r < OW; r += 8) {
        const float* row0 = &s_img[r * HW];
        const float* row1 = row0 + HW;
        for (int c = lane; c < OW; c += 32) {
            float x = w00 * row0[c] + w01 * row0[c + 1]
                    + w10 * row1[c] + w11 * row1[c + 1] + bias;
            sum += 1.0f / (1.0f + __expf(-x));
        }
    }

    // ---- wave reduction via V_WMMA_F32_16X16X4_F32 ----
    // A layout (16x4 f32): lanes 0-15 -> {K=0,K=1}, lanes 16-31 -> {K=2,K=3}.
    // A = {sum, 0} per lane; B = ones  =>  D[m][n] = s_m + s_{m+16}.
    v2f a;    a[0] = sum;  a[1] = 0.0f;
    v2f ones; ones[0] = 1.0f; ones[1] = 1.0f;
    v8f acc = {};
    acc = __builtin_amdgcn_wmma_f32_16x16x4_f32(
        /*neg_a=*/false, a, /*neg_b=*/false, ones,
        /*c_mod=*/(short)0, acc, /*reuse_a=*/false, /*reuse_b=*/false);
    // lane n<16 holds D[0..7][n]; lane n+16 holds D[8..15][n]
    float t = acc[0] + acc[1] + acc[2] + acc[3]
            + acc[4] + acc[5] + acc[6] + acc[7];
    t += __shfl_xor(t, 16);          // total wave sum in every lane

    // ---- fold 8 wave sums (overlay LDS after all reads are done) ----
    __syncthreads();                  // everyone finished reading s_img
    if (lane == 0) s_img[wave] = t;
    __syncthreads();
    if (tid == 0) {
        float tot = 0.0f;
#pragma unroll
        for (int i = 0; i < 8; ++i) tot += s_img[i];
        means[b] = tot * INV_OUT_ELEMS;   // mean without IEEE divide
    }
}

// Single block: tiny 4-layer MLP on [nrows,2], final linear, RBF scalar.
__global__ __launch_bounds__(256) void fraud_tail_kernel(
    const float* __restrict__ means,
    const float* __restrict__ fraud_W, const float* __restrict__ fraud_b,
    const float* __restrict__ scale,   const float* __restrict__ shift,
    const float* __restrict__ final_W, const float* __restrict__ final_b,
    float* __restrict__ out, int nrows)
{
    __shared__ float s_red[256];
    const int tid = threadIdx.x;

    const float fw0 = final_W[0], fw1 = final_W[1], fb = final_b[0];

    float ss = 0.0f;
    for (int i = tid; i < nrows; i += 256) {
        float a0 = means[2 * i + 0];
        float a1 = means[2 * i + 1];
#pragma unroll
        for (int l = 0; l < NLAYER; ++l) {
            const float* W = fraud_W + l * 4;      // [l][2][2] row-major
            float n0 = tanhf(a0 * W[0] + a1 * W[1] + fraud_b[l * 2 + 0])
                     * scale[l * 2 + 0] + shift[l * 2 + 0];
            float n1 = tanhf(a0 * W[2] + a1 * W[3] + fraud_b[l * 2 + 1])
                     * scale[l * 2 + 1] + shift[l * 2 + 1];
            a0 = n0; a1 = n1;
        }
        float y = a0 * fw0 + a1 * fw1 + fb;        // x @ final_W.T + b
        ss += y * y;
    }

    s_red[tid] = ss;
    __syncthreads();
#pragma unroll
    for (int s = 128; s > 0; s >>= 1) {
        if (tid < s) s_red[tid] += s_red[tid + s];
        __syncthreads();
    }
    if (tid == 0) out[0] = __expf(-GAMMA_F * s_red[0]);
}

extern "C" void kernel_launch(void* const* d_in, const int* in_sizes, int n_in,
                              void* d_out, int out_size, void* d_ws, size_t ws_size,
                              hipStream_t stream) {
    // inputs (setup_inputs order):
    // 0: data [B,1,128,128]  1: conv_w [1,1,2,2]  2: conv_b [1]
    // 3: fraud_W [4,2,2]     4: fraud_b [4,2]     5: scale [4,2]
    // 6: shift [4,2]         7: final_W [1,2]     8: final_b [1]
    const float* data    = (const float*)d_in[0];
    const float* conv_w  = (const float*)d_in[1];
    const float* conv_b  = (const float*)d_in[2];
    const float* fraud_W = (const float*)d_in[3];
    const float* fraud_b = (const float*)d_in[4];
    const float* scale   = (const float*)d_in[5];
    const float* shift   = (const float*)d_in[6];
    const float* final_W = (const float*)d_in[7];
    const float* final_b = (const float*)d_in[8];
    float* out = (float*)d_out;

    const int B = in_sizes[0] / IMG_ELEMS;   // 4096
    float* means = (float*)d_ws;             // B floats of scratch

    conv_sig_mean_kernel<<<B, 256, 0, stream>>>(data, conv_w, conv_b, means);
    fraud_tail_kernel<<<1, 256, 0, stream>>>(means, fraud_W, fraud_b, scale,
                                             shift, final_W, final_b, out, B / 2);
}